// QState_64278480552187
// MI455X (gfx1250) — compile-verified
//
#include <hip/hip_runtime.h>

// ===========================================================================
// 20-qubit tensor-product gate simulation as two complex 1024^3 GEMMs per
// batch, executed with V_WMMA_F32_16X16X4_F32.  out_b = A_b * M * B_b^T,
// A_b = Ahi (x) Alo, B_b = Bhi (x) Blo  (32x32 complex Kronecker factors,
// generated on the fly from LDS -> WMMA A-fragments; no big operand in HBM).
// ===========================================================================

static constexpr int QL = 1024;            // matrix dim = 2^10
static constexpr unsigned QDIM = 1u << 20; // state size

typedef float v2f __attribute__((ext_vector_type(2)));
typedef float v8f __attribute__((ext_vector_type(8)));

__device__ __forceinline__ v8f wmma4(v2f a, v2f b, v8f c) {
  // D = A(16x4 f32) x B(4x16 f32) + C(16x16 f32)
  return __builtin_amdgcn_wmma_f32_16x16x4_f32(false, a, false, b, (short)0, c,
                                               false, false);
}

#define QC 0.70710678118654752440f
__constant__ float kGR[3][2][2] = {
    {{QC, QC}, {QC, -QC}}, {{0.f, QC}, {QC, 0.f}}, {{1.f, 0.f}, {0.f, 1.f}}};
__constant__ float kGI[3][2][2] = {
    {{0.f, 0.f}, {0.f, 0.f}}, {{-QC, 0.f}, {0.f, -QC}}, {{0.f, 0.f}, {0.f, 0.f}}};

// ---------------------------------------------------------------------------
// Transpose the (batch-independent) initial state M (1024x1024, re/im planes)
// into Mt so GEMM-1 can load its B-operand fragments coalesced.
// ---------------------------------------------------------------------------
__global__ __launch_bounds__(256) void qs_transpose(
    const float* __restrict__ sre, const float* __restrict__ sim,
    float* __restrict__ mtre, float* __restrict__ mtim) {
  __shared__ float tile[2][32][33];
  const int t = threadIdx.x;
  const int row = t >> 3;
  const int c4 = (t & 7) << 2;
  const int bx = blockIdx.x, by = blockIdx.y;
  const float4 r4 = *(const float4*)&sre[(by * 32 + row) * QL + bx * 32 + c4];
  const float4 i4 = *(const float4*)&sim[(by * 32 + row) * QL + bx * 32 + c4];
  tile[0][row][c4 + 0] = r4.x; tile[0][row][c4 + 1] = r4.y;
  tile[0][row][c4 + 2] = r4.z; tile[0][row][c4 + 3] = r4.w;
  tile[1][row][c4 + 0] = i4.x; tile[1][row][c4 + 1] = i4.y;
  tile[1][row][c4 + 2] = i4.z; tile[1][row][c4 + 3] = i4.w;
  __syncthreads();
  float4 o;
  o.x = tile[0][c4 + 0][row]; o.y = tile[0][c4 + 1][row];
  o.z = tile[0][c4 + 2][row]; o.w = tile[0][c4 + 3][row];
  *(float4*)&mtre[(bx * 32 + row) * QL + by * 32 + c4] = o;
  o.x = tile[1][c4 + 0][row]; o.y = tile[1][c4 + 1][row];
  o.z = tile[1][c4 + 2][row]; o.w = tile[1][c4 + 3][row];
  *(float4*)&mtim[(bx * 32 + row) * QL + by * 32 + c4] = o;
}

// ---------------------------------------------------------------------------
// Per batch build four 32x32 complex Kronecker factors (5 gates each).
// F layout: [(b*4 + m)*2 + {re,im}][1024], m: 0=Ahi 1=Alo 2=Bhi 3=Blo.
// Matrix m entry [u,v] = prod_{j=0..4} C4[obs[b, 4+5m-j]][u_j, v_j].
// ---------------------------------------------------------------------------
__global__ __launch_bounds__(256) void qs_factors(const int* __restrict__ obs,
                                                  float* __restrict__ F) {
  const int b = blockIdx.x;
  for (int e = threadIdx.x; e < 4096; e += 256) {
    const int m = e >> 10, idx = e & 1023, u = idx >> 5, v = idx & 31;
    float pr = 1.f, pi = 0.f;
#pragma unroll
    for (int j = 0; j < 5; ++j) {
      const int g = obs[b * 20 + (4 + 5 * m - j)];
      const int ub = (u >> j) & 1, vb = (v >> j) & 1;
      const float gr = kGR[g][ub][vb], gi = kGI[g][ub][vb];
      const float nr = pr * gr - pi * gi;
      const float ni = pr * gi + pi * gr;
      pr = nr; pi = ni;
    }
    F[(long long)(b * 8 + m * 2 + 0) * 1024 + idx] = pr;
    F[(long long)(b * 8 + m * 2 + 1) * 1024 + idx] = pi;
  }
}

// ---------------------------------------------------------------------------
// Generic complex GEMM:  Y[r,c] = sum_k W[r,k] * X[k,c],
//   W = Whi (x) Wlo generated from LDS as WMMA A-fragments,
//   X loaded either directly (row-major [k][c], coalesced)      [XT=false]
//   or staged through padded LDS from a transposed source [c][k] [XT=true].
// Output row-major; PROBS writes |Y|^2, else re/im planes.
// Workgroup = 256 thr (8 waves) -> 128x64 output tile; wave = 2x2 16x16 tiles.
// ---------------------------------------------------------------------------
template <bool XT, bool PROBS>
__global__ __launch_bounds__(256) void qs_gemm(
    const float* __restrict__ Xbase, long long xPerZ,
    const float* __restrict__ F, int facOff, int b0,
    float* __restrict__ outBase, long long outPerZ, int outZ0) {
  __shared__ float sWhiRe[32 * 34];
  __shared__ float sWhiIm[32 * 34];
  __shared__ float sWloRe[32 * 34];
  __shared__ float sWloIm[32 * 34];
  __shared__ float sXre[64 * 34];
  __shared__ float sXim[64 * 34];

  const int z = blockIdx.z;
  const float* __restrict__ Xre = Xbase + (long long)z * xPerZ;
  const float* __restrict__ Xim = Xre + QDIM;
  const float* __restrict__ fac = F + (long long)(b0 + z) * 8192 + facOff;
  float* __restrict__ outRe = outBase + (long long)(z + outZ0) * outPerZ;
  float* __restrict__ outIm = outRe + QDIM;

  // stage the four 32x32 factor planes into padded LDS
  for (int e = threadIdx.x; e < 4096; e += 256) {
    const int pl = e >> 10, idx = e & 1023, u = idx >> 5, vv = idx & 31;
    float* dst = (pl == 0) ? sWhiRe : (pl == 1) ? sWhiIm
                 : (pl == 2) ? sWloRe : sWloIm;
    dst[u * 34 + vv] = fac[e];
  }
  __syncthreads();

  const int lane = threadIdx.x & 31;
  const int wv = threadIdx.x >> 5;
  const int wrow = wv & 3;          // 4 row groups
  const int wcol = wv >> 2;         // 2 col groups
  const int lhi = lane >> 4;
  const int llo = lane & 15;
  const int rb = blockIdx.y * 128 + wrow * 32;  // multiple of 32
  const int cblk = blockIdx.x * 64;
  const int cb = cblk + wcol * 32;

  const v8f vzero = {0.f, 0.f, 0.f, 0.f, 0.f, 0.f, 0.f, 0.f};
  v8f acc[2][2][3];
#pragma unroll
  for (int a = 0; a < 2; ++a)
#pragma unroll
    for (int b = 0; b < 2; ++b)
#pragma unroll
      for (int c = 0; c < 3; ++c) acc[a][b][c] = vzero;

  const int ruW = rb >> 5;          // Whi row (shared by both row tiles)
  const int rl0 = llo;              // Wlo rows for row tiles 0/1
  const int rl1 = 16 + llo;

  for (int k32 = 0; k32 < 32; ++k32) {
    const int kbase = k32 << 5;
    if constexpr (XT) {
      __syncthreads();
      // stage [64 c][32 k] from source stored [c][k] (rows of X^T), coalesced
      const int cc = threadIdx.x >> 2;
      const int kk = (threadIdx.x & 3) << 3;
      const long long so = (long long)(cblk + cc) * QL + kbase + kk;
      const float4 xr0 = *(const float4*)&Xre[so];
      const float4 xr1 = *(const float4*)&Xre[so + 4];
      const float4 xi0 = *(const float4*)&Xim[so];
      const float4 xi1 = *(const float4*)&Xim[so + 4];
      float* dr = &sXre[cc * 34 + kk];
      dr[0] = xr0.x; dr[1] = xr0.y; dr[2] = xr0.z; dr[3] = xr0.w;
      dr[4] = xr1.x; dr[5] = xr1.y; dr[6] = xr1.z; dr[7] = xr1.w;
      float* di = &sXim[cc * 34 + kk];
      di[0] = xi0.x; di[1] = xi0.y; di[2] = xi0.z; di[3] = xi0.w;
      di[4] = xi1.x; di[5] = xi1.y; di[6] = xi1.z; di[7] = xi1.w;
      __syncthreads();
    }
    const float whiRe = sWhiRe[ruW * 34 + k32];
    const float whiIm = sWhiIm[ruW * 34 + k32];
#pragma unroll
    for (int ks = 0; ks < 8; ++ks) {
      const int kloc = (ks << 2) + (lhi << 1);  // even -> b64 LDS loads
      // --- generated A-fragments: W[r,k] = whi * Wlo[r&31, k&31] ---
      const float2 lr0 = *(const float2*)&sWloRe[rl0 * 34 + kloc];
      const float2 li0 = *(const float2*)&sWloIm[rl0 * 34 + kloc];
      const float2 lr1 = *(const float2*)&sWloRe[rl1 * 34 + kloc];
      const float2 li1 = *(const float2*)&sWloIm[rl1 * 34 + kloc];
      v2f aRe[2], aIm[2];
      aRe[0].x = whiRe * lr0.x - whiIm * li0.x;
      aRe[0].y = whiRe * lr0.y - whiIm * li0.y;
      aIm[0].x = whiRe * li0.x + whiIm * lr0.x;
      aIm[0].y = whiRe * li0.y + whiIm * lr0.y;
      aRe[1].x = whiRe * lr1.x - whiIm * li1.x;
      aRe[1].y = whiRe * lr1.y - whiIm * li1.y;
      aIm[1].x = whiRe * li1.x + whiIm * lr1.x;
      aIm[1].y = whiRe * li1.y + whiIm * lr1.y;
      // --- B-fragments: lane holds (K = v + 2*lhi, N = llo) ---
      v2f bRe[2], bIm[2];
      if constexpr (XT) {
        const int cl0 = wcol * 32 + llo;
        const int cl1 = cl0 + 16;
        float2 t;
        t = *(const float2*)&sXre[cl0 * 34 + kloc]; bRe[0].x = t.x; bRe[0].y = t.y;
        t = *(const float2*)&sXim[cl0 * 34 + kloc]; bIm[0].x = t.x; bIm[0].y = t.y;
        t = *(const float2*)&sXre[cl1 * 34 + kloc]; bRe[1].x = t.x; bRe[1].y = t.y;
        t = *(const float2*)&sXim[cl1 * 34 + kloc]; bIm[1].x = t.x; bIm[1].y = t.y;
      } else {
        const int krow = kbase + (ks << 2) + (lhi << 1);
        const int c0 = cb + llo;
        const int c1 = c0 + 16;
        bRe[0].x = Xre[krow * QL + c0]; bRe[0].y = Xre[(krow + 1) * QL + c0];
        bIm[0].x = Xim[krow * QL + c0]; bIm[0].y = Xim[(krow + 1) * QL + c0];
        bRe[1].x = Xre[krow * QL + c1]; bRe[1].y = Xre[(krow + 1) * QL + c1];
        bIm[1].x = Xim[krow * QL + c1]; bIm[1].y = Xim[(krow + 1) * QL + c1];
      }
      // --- complex MAC: 4 WMMAs per 16x16 tile per k-step ---
#pragma unroll
      for (int rt = 0; rt < 2; ++rt)
#pragma unroll
        for (int ct = 0; ct < 2; ++ct) {
          acc[rt][ct][0] = wmma4(aRe[rt], bRe[ct], acc[rt][ct][0]);
          acc[rt][ct][1] = wmma4(aIm[rt], bIm[ct], acc[rt][ct][1]);
          acc[rt][ct][2] = wmma4(aRe[rt], bIm[ct], acc[rt][ct][2]);
          acc[rt][ct][2] = wmma4(aIm[rt], bRe[ct], acc[rt][ct][2]);
        }
    }
  }

  // --- store (row-major, coalesced 64B half-wave segments) ---
#pragma unroll
  for (int rt = 0; rt < 2; ++rt) {
#pragma unroll
    for (int ct = 0; ct < 2; ++ct) {
      const v8f yre = acc[rt][ct][0] - acc[rt][ct][1];
      const v8f yim = acc[rt][ct][2];
      const int r0 = rb + rt * 16 + 8 * lhi;
      const int col = cb + ct * 16 + llo;
#pragma unroll
      for (int j = 0; j < 8; ++j) {
        const int row = r0 + j;
        if constexpr (PROBS) {
          outRe[row * QL + col] = yre[j] * yre[j] + yim[j] * yim[j];
        } else {
          outRe[row * QL + col] = yre[j];
          outIm[row * QL + col] = yim[j];
        }
      }
    }
  }
}

// ---------------------------------------------------------------------------
extern "C" void kernel_launch(void* const* d_in, const int* in_sizes, int n_in,
                              void* d_out, int out_size, void* d_ws,
                              size_t ws_size, hipStream_t stream) {
  (void)in_sizes; (void)n_in; (void)out_size;
  const float* sre = (const float*)d_in[0];
  const float* sim = (const float*)d_in[1];
  const int* obs = (const int*)d_in[2];
  float* probs = (float*)d_out;
  float* ws = (float*)d_ws;

  const long long DIMf = (long long)QDIM;
  float* F = ws;                       // 32 * 8192 floats = 1 MB
  float* Mt = ws + 32 * 8192;          // 2 * DIM floats   = 8 MB
  float* T = Mt + 2 * DIMf;            // chunk * 2 * DIM floats

  // deterministic batch chunking based on available workspace
  const long long wsFloats = (long long)(ws_size / 4);
  long long avail = wsFloats - (32 * 8192 + 2 * DIMf);
  if (avail < 0) avail = 0;
  int chunk = (int)(avail / (2 * DIMf));
  if (chunk < 1) chunk = 1;
  if (chunk > 32) chunk = 32;

  qs_transpose<<<dim3(32, 32), 256, 0, stream>>>(sre, sim, Mt, Mt + DIMf);
  qs_factors<<<dim3(32), 256, 0, stream>>>(obs, F);

  for (int b0 = 0; b0 < 32; b0 += chunk) {
    const int n = (32 - b0 < chunk) ? (32 - b0) : chunk;
    // GEMM-1: Tt[xl,xh] = B * M^T   (B-factors, direct coalesced X loads)
    qs_gemm<false, false><<<dim3(16, 8, n), 256, 0, stream>>>(
        Mt, 0LL, F, 4096, b0, T, 2 * DIMf, 0);
    // GEMM-2: out[xh,xl] = A * T    (A-factors, LDS-staged transposed X),
    //         writes probs = |out|^2
    qs_gemm<true, true><<<dim3(16, 8, n), 256, 0, stream>>>(
        T, 2 * DIMf, F, 0, b0, probs, DIMf, b0);
  }
}